// NewRobertaLayerNorm_40278203302412
// MI455X (gfx1250) — compile-verified
//
#include <hip/hip_runtime.h>
#include <cstdint>

#define HIDDEN   768
#define NUM_SEG  64
#define EPS      1e-5f
#define THREADS  192   // 6 wave32s; 192 * float4 = 768 floats = one row
#define NWAVES   (THREADS / 32)

#if defined(__HIP_DEVICE_COMPILE__)
typedef int v4i __attribute__((ext_vector_type(4)));
typedef __attribute__((address_space(1))) v4i gv4i_t;   // global int4
typedef __attribute__((address_space(3))) v4i lv4i_t;   // LDS int4
typedef __attribute__((address_space(3))) void lvoid_t;

__device__ static inline unsigned lds_byte_addr(void* p) {
    // generic -> LDS addrspace -> 32-bit LDS byte offset
    return (unsigned)(unsigned long long)(lvoid_t*)p;
}
#endif

__global__ __launch_bounds__(THREADS) void rln_kernel(
    const float* __restrict__ x,
    const float* __restrict__ weight,
    const float* __restrict__ bias,
    const float* __restrict__ seg_x,
    const float* __restrict__ seg_y,
    float* __restrict__ out)
{
    __shared__ float tile[HIDDEN];
    __shared__ float part_s[NWAVES];
    __shared__ float part_q[NWAVES];
    __shared__ float s_mean;
    __shared__ float s_inv;

    const int t = (int)threadIdx.x;
    const long long row = (long long)blockIdx.x;
    const float* __restrict__ xrow = x   + row * (long long)HIDDEN;
    float*       __restrict__ orow = out + row * (long long)HIDDEN;

    // ---- Stage one row (3 KB) into LDS via CDNA5 async global->LDS copy ----
#if defined(__HIP_DEVICE_COMPILE__) && defined(__gfx1250__)
  #if __has_builtin(__builtin_amdgcn_global_load_async_to_lds_b128)
    __builtin_amdgcn_global_load_async_to_lds_b128(
        (gv4i_t*)(xrow + 4 * t), (lv4i_t*)(&tile[4 * t]), 0, 0);
  #else
    {
        unsigned lds_dst = lds_byte_addr(&tile[4 * t]);
        const float* gsrc = xrow + 4 * t;
        asm volatile("global_load_async_to_lds_b128 %0, %1, off"
                     :
                     : "v"(lds_dst), "v"(gsrc)
                     : "memory");
    }
  #endif
  #if __has_builtin(__builtin_amdgcn_s_wait_asynccnt)
    __builtin_amdgcn_s_wait_asynccnt(0);
  #else
    asm volatile("s_wait_asynccnt 0" ::: "memory");
  #endif
#else
    // host pass / non-gfx1250 fallback: plain vector load through registers
    *(float4*)(&tile[4 * t]) = *(const float4*)(xrow + 4 * t);
#endif
    __syncthreads();

    // ---- Pull own float4 from LDS, accumulate sum / sum-of-squares ----
    float4 v = *(const float4*)(&tile[4 * t]);
    float s = v.x + v.y + v.z + v.w;
    float q = v.x * v.x + v.y * v.y + v.z * v.z + v.w * v.w;

    // wave32 butterfly reduction
#pragma unroll
    for (int off = 16; off > 0; off >>= 1) {
        s += __shfl_xor(s, off, 32);
        q += __shfl_xor(q, off, 32);
    }
    const int wave = t >> 5;
    const int lane = t & 31;
    if (lane == 0) { part_s[wave] = s; part_q[wave] = q; }
    __syncthreads();

    // ---- Final reduce + piecewise-linear 1/sqrt(var+eps) on thread 0 ----
    if (t == 0) {
        float S = 0.f, Q = 0.f;
#pragma unroll
        for (int w = 0; w < NWAVES; ++w) { S += part_s[w]; Q += part_q[w]; }
        const float invH = 1.0f / (float)HIDDEN;
        const float mean = S * invH;
        float var = Q * invH - mean * mean;
        float vv  = var + EPS;

        const float lo = seg_x[0];
        const float hi = seg_x[NUM_SEG];
        float inv;
        if (vv <= lo) {
            inv = seg_y[0];
        } else if (vv >= hi) {
            inv = seg_y[NUM_SEG];
        } else {
            // breakpoints are log-spaced: analytic index guess, then fixup
            float fr = (__log2f(vv) - __log2f(lo)) *
                       ((float)NUM_SEG / (__log2f(hi) - __log2f(lo)));
            int i = (int)fr;
            if (i < 0) i = 0;
            if (i > NUM_SEG - 1) i = NUM_SEG - 1;
            while (i > 0 && vv < seg_x[i]) --i;
            while (i < NUM_SEG - 1 && vv >= seg_x[i + 1]) ++i;
            const float x0 = seg_x[i], x1 = seg_x[i + 1];
            const float tt = (vv - x0) / (x1 - x0);
            inv = seg_y[i] + (seg_y[i + 1] - seg_y[i]) * tt;
        }
        s_mean = mean;
        s_inv  = inv;
    }
    __syncthreads();

    // ---- Normalize + affine, b128 store ----
    const float mean = s_mean;
    const float inv  = s_inv;
    const float4 w4  = *(const float4*)(weight + 4 * t);
    const float4 b4  = *(const float4*)(bias   + 4 * t);
    float4 o;
    o.x = (v.x - mean) * inv * w4.x + b4.x;
    o.y = (v.y - mean) * inv * w4.y + b4.y;
    o.z = (v.z - mean) * inv * w4.z + b4.z;
    o.w = (v.w - mean) * inv * w4.w + b4.w;
    *(float4*)(orow + 4 * t) = o;
}

extern "C" void kernel_launch(void* const* d_in, const int* in_sizes, int n_in,
                              void* d_out, int out_size, void* d_ws, size_t ws_size,
                              hipStream_t stream) {
    const float* x  = (const float*)d_in[0];
    const float* w  = (const float*)d_in[1];
    const float* b  = (const float*)d_in[2];
    const float* sx = (const float*)d_in[3];
    const float* sy = (const float*)d_in[4];
    float* out = (float*)d_out;

    const int rows = in_sizes[0] / HIDDEN;   // 8 * 4096 = 32768
    dim3 grid((unsigned)rows), block(THREADS);
    hipLaunchKernelGGL(rln_kernel, grid, block, 0, stream, x, w, b, sx, sy, out);
}